// DSVTInputLayer_20091857011192
// MI455X (gfx1250) — compile-verified
//
#include <hip/hip_runtime.h>
#include <hip/hip_bf16.h>

// ---------------------------------------------------------------------------
// DSVT input layer for MI455X (gfx1250, wave32).
//  - set partitioning via window-cell scatter + compaction (no sorts needed)
//  - BN stats via analytic moments of loc (5 scalars per shift)
//  - pos-embed second linear as fp32 WMMA (v_wmma_f32_16x16x4_f32),
//    W2 column panel staged to LDS by the Tensor Data Mover (TDM).
// ---------------------------------------------------------------------------

typedef float v2f  __attribute__((ext_vector_type(2)));
typedef float v8f  __attribute__((ext_vector_type(8)));
typedef unsigned int u32x4 __attribute__((ext_vector_type(4)));
typedef int   i32x4 __attribute__((ext_vector_type(4)));
typedef int   i32x8 __attribute__((ext_vector_type(8)));

#define DIMC   192
#define SETSZ  36
#define NW0    6400   // 2 * 40*40*2 windows, shift 0 (12x12 win)
#define MV0    144    // 12*12*1
#define NW1    1764   // 2 * 21*21*2 windows, shift 1 (24x24 win)
#define MV1    576    // 24*24*1

// ---------------- window scatter: win counts, cell scatter, locs -----------
__global__ void k_win(const int* __restrict__ coords, float* __restrict__ locs,
                      int* __restrict__ cnt0, int* __restrict__ cnt1,
                      int* __restrict__ c0y, int* __restrict__ c0x,
                      int* __restrict__ c1y, int* __restrict__ c1x, int N)
{
    int i = blockIdx.x * blockDim.x + threadIdx.x;
    if (i >= N) return;
    int b = coords[4 * i + 0];
    int y = coords[4 * i + 2];
    int x = coords[4 * i + 3];
    // shift 0: win 12x12, mx=my=40, mz=2  -> w = b*3200 + (x/12)*80 + (y/12)*2
    {
        int w = b * 3200 + (x / 12) * 80 + (y / 12) * 2;
        atomicAdd(&cnt0[w], 1);
        int xin = x % 12, yin = y % 12;
        c0y[w * MV0 + yin * 12 + xin] = i;   // unique cell -> plain store
        c0x[w * MV0 + xin * 12 + yin] = i;
        locs[2 * i + 0] = (float)xin - 6.0f;
        locs[2 * i + 1] = (float)yin - 6.0f;
    }
    // shift 1: win 24x24 shifted by (6,6), mx=my=21, mz=2
    {
        int sx = x + 6, sy = y + 6;
        int w = b * 882 + (sx / 24) * 42 + (sy / 24) * 2;
        atomicAdd(&cnt1[w], 1);
        int xin = sx % 24, yin = sy % 24;
        c1y[w * MV1 + yin * 24 + xin] = i;
        c1x[w * MV1 + xin * 24 + yin] = i;
        locs[2 * (N + i) + 0] = (float)xin - 6.0f;  // pos-embed uses base win/2 = 6
        locs[2 * (N + i) + 1] = (float)yin - 6.0f;
    }
}

// ---------------- loc moment partials (deterministic tree reduce) ----------
__global__ void k_stats_part(const float* __restrict__ locs, float* __restrict__ part,
                             int N, int nblk)
{
    __shared__ float red[256];
    int i = blockIdx.x * 256 + threadIdx.x;
    for (int s = 0; s < 2; ++s) {
        float lx = 0.0f, ly = 0.0f;
        if (i < N) { lx = locs[2 * ((size_t)s * N + i) + 0]; ly = locs[2 * ((size_t)s * N + i) + 1]; }
        float v[5] = { lx, ly, lx * lx, ly * ly, lx * ly };
        for (int q = 0; q < 5; ++q) {
            red[threadIdx.x] = v[q];
            __syncthreads();
            for (int off = 128; off > 0; off >>= 1) {
                if (threadIdx.x < off) red[threadIdx.x] += red[threadIdx.x + off];
                __syncthreads();
            }
            if (threadIdx.x == 0) part[(size_t)(s * 5 + q) * nblk + blockIdx.x] = red[0];
            __syncthreads();
        }
    }
}

__global__ void k_stats_final(const float* __restrict__ part, float* __restrict__ statf, int nblk)
{
    int t = threadIdx.x;
    if (t >= 10) return;
    float a = 0.0f;
    for (int j = 0; j < nblk; ++j) a += part[(size_t)t * nblk + j];
    statf[t] = a;
}

// ---------- fold Linear1 + BN(batch stats) + ReLU into per-channel affine --
__global__ void k_params(const float* __restrict__ statf, const float* __restrict__ w1,
                         const float* __restrict__ b1, const float* __restrict__ gamma,
                         const float* __restrict__ beta, float* __restrict__ pqr, int N)
{
    int t = blockIdx.x * blockDim.x + threadIdx.x;
    if (t >= 8 * DIMC) return;
    int k = t / DIMC, c = t % DIMC;
    int s = k & 1;
    float inv = 1.0f / (float)N;
    float mx = statf[s * 5 + 0] * inv, my = statf[s * 5 + 1] * inv;
    float vxx = statf[s * 5 + 2] * inv - mx * mx;
    float vyy = statf[s * 5 + 3] * inv - my * my;
    float vxy = statf[s * 5 + 4] * inv - mx * my;
    float a  = w1[(k * 2 + 0) * DIMC + c];
    float bb = w1[(k * 2 + 1) * DIMC + c];
    float bias = b1[k * DIMC + c];
    float mu  = mx * a + my * bb + bias;
    float var = a * a * vxx + 2.0f * a * bb * vxy + bb * bb * vyy;
    float sc  = gamma[k * DIMC + c] * rsqrtf(var + 1e-5f);
    pqr[(k * 3 + 0) * DIMC + c] = sc * a;
    pqr[(k * 3 + 1) * DIMC + c] = sc * bb;
    pqr[(k * 3 + 2) * DIMC + c] = sc * bias + beta[k * DIMC + c] - mu * sc;
}

// ---------------- set-offset scans (serial => deterministic, tiny) ---------
__global__ void k_scan(const int* __restrict__ cnt0, int* __restrict__ off0,
                       const int* __restrict__ cnt1, int* __restrict__ off1,
                       int* __restrict__ S)
{
    int t = threadIdx.x;
    if (t == 0) { int acc = 0; for (int w = 0; w < NW0; ++w) { off0[w] = acc; acc += (cnt0[w] + SETSZ - 1) / SETSZ; } S[0] = acc; }
    if (t == 1) { int acc = 0; for (int w = 0; w < NW1; ++w) { off1[w] = acc; acc += (cnt1[w] + SETSZ - 1) / SETSZ; } S[1] = acc; }
}

// -------- per-window in-place compaction -> spatial-rank ordered voxels ----
__global__ void k_compact(int* __restrict__ c0y, int* __restrict__ c0x,
                          int* __restrict__ c1y, int* __restrict__ c1x)
{
    int t = blockIdx.x * blockDim.x + threadIdx.x;
    int* cell; int mv; int w;
    if      (t < NW0)                 { cell = c0y; mv = MV0; w = t; }
    else if (t < 2 * NW0)             { cell = c0x; mv = MV0; w = t - NW0; }
    else if (t < 2 * NW0 + NW1)       { cell = c1y; mv = MV1; w = t - 2 * NW0; }
    else if (t < 2 * NW0 + 2 * NW1)   { cell = c1x; mv = MV1; w = t - 2 * NW0 - NW1; }
    else return;
    long base = (long)w * mv;
    int m = 0;
    for (int c = 0; c < mv; ++c) {
        int v = cell[base + c];
        if (v >= 0) { cell[base + m] = v; ++m; }
    }
}

// ---------------- emit set_voxel_inds + masks ------------------------------
__global__ void k_emit(const int* __restrict__ cnt, const int* __restrict__ off,
                       const int* __restrict__ cy, const int* __restrict__ cx,
                       const int* __restrict__ S, float* __restrict__ out,
                       long base0, int mv, int shift)
{
    int w = blockIdx.x;
    int c = cnt[w];
    if (c <= 0) return;
    int setnum = (c + SETSZ - 1) / SETSZ;
    int t = threadIdx.x;
    int s = t / SETSZ, e = t % SETSZ;
    if (s >= setnum) return;
    long Sx = S[shift];
    long base = base0 + (shift ? 4L * (long)S[0] * SETSZ : 0L);
    long g = (long)off[w] + s;
    int den = setnum * SETSZ;
    int idx  = (int)(((long)(s * SETSZ + e) * c) / den);        // DSVT Eq.3 spread
    int idxp = e ? (int)(((long)(s * SETSZ + e - 1) * c) / den) : -1;
    long maskBase = base + 2L * Sx * SETSZ;
    const int* cells[2] = { cy, cx };
    for (int o = 0; o < 2; ++o) {
        int vox = cells[o][(long)w * mv + idx];
        int pv  = e ? cells[o][(long)w * mv + idxp] : -1;
        ((int*)out)[base + ((long)o * Sx + g) * SETSZ + e] = vox;       // int32 bits
        out[maskBase + ((long)o * Sx + g) * SETSZ + e] = (vox == pv) ? 1.0f : 0.0f;
    }
}

// ---------------- pos-embed GEMM: relu(affine(loc)) @ W2 + b2 --------------
// fp32 WMMA 16x16x4; W2 column panel (192x16, 12KB) staged to LDS by TDM.
__global__ __launch_bounds__(256)
void k_posembed(const float* __restrict__ locs, const float* __restrict__ pqr,
                const float* __restrict__ w2, const float* __restrict__ b2,
                float* __restrict__ out, long posBase, int N)
{
    __shared__ float lds_w2[DIMC * 16];   // MUST be first shared object: TDM D#.lds_addr = 0
    __shared__ float lds_pqr[3 * DIMC];

    const int k    = blockIdx.z;          // 0..7 (block x shift embedding)
    const int j0   = blockIdx.y;          // 0..11 column tile
    const int rc   = blockIdx.x;          // row chunk (8 waves * 16 rows)
    const int tid  = threadIdx.x;
    const int wave = tid >> 5;
    const int lane = tid & 31;

    if (wave == 0) {
        // Tensor DMA descriptor (D#): 2D tensor 192x192 fp32, tile 16 (dim0) x 192 (dim1)
        unsigned long long ga =
            (unsigned long long)(const void*)(w2 + (size_t)k * DIMC * DIMC + (size_t)j0 * 16);
        u32x4 g0;
        g0[0] = 1u;                                   // count=1, user mode, no gather
        g0[1] = 0u;                                   // lds_addr = 0 (lds_w2)
        g0[2] = (unsigned)ga;                         // global_addr[31:0]
        g0[3] = ((unsigned)(ga >> 32) & 0x01FFFFFFu) | 0x80000000u; // addr[56:32], type=2
        i32x8 g1;
        g1[0] = 2 << 16;                              // data_size = 4 bytes
        g1[1] = DIMC << 16;                           // tensor_dim0[15:0]
        g1[2] = DIMC << 16;                           // tensor_dim1[15:0]
        g1[3] = 16 << 16;                             // tile_dim0 = 16
        g1[4] = DIMC;                                 // tile_dim1 = 192
        g1[5] = DIMC;                                 // tensor_dim0_stride = 192
        g1[6] = 0;
        g1[7] = 0;
        i32x4 zz; zz[0] = zz[1] = zz[2] = zz[3] = 0;
        i32x8 z8; z8[0] = z8[1] = z8[2] = z8[3] = z8[4] = z8[5] = z8[6] = z8[7] = 0;
        __builtin_amdgcn_tensor_load_to_lds(g0, g1, zz, zz, z8, 0);   // 6-arg (clang-23) form
        __builtin_amdgcn_s_wait_tensorcnt(0);
    }
    // The TDM writes lds_w2 "behind the compiler's back" (the LDS address is an
    // integer inside the descriptor). Escape the pointer into an asm with a
    // memory clobber so the optimizer cannot treat lds_w2 as never-written and
    // fold the B-fragment loads to undef (observed in round-2 disasm).
    {
        float* p = lds_w2;
        asm volatile("" : "+v"(p) : : "memory");
    }
    for (int i = tid; i < 3 * DIMC; i += 256)
        lds_pqr[i] = pqr[(size_t)k * 3 * DIMC + i];
    __syncthreads();

    const int rowBase = (rc * 8 + wave) * 16;
    if (rowBase >= N) return;                         // wave-uniform (N % 16 == 0)

    const int m    = lane & 15;                       // row within tile / out column
    const int half = lane >> 4;
    const int row  = rowBase + m;
    const int shift = k & 1;
    const float lx = locs[2 * ((size_t)shift * N + row) + 0];
    const float ly = locs[2 * ((size_t)shift * N + row) + 1];
    const int col  = j0 * 16 + m;
    const float b2v = b2[k * DIMC + col];

    v8f acc;
#pragma unroll
    for (int i = 0; i < 8; ++i) acc[i] = b2v;

    const float* P = lds_pqr;
    const float* Q = lds_pqr + DIMC;
    const float* R = lds_pqr + 2 * DIMC;

#pragma unroll
    for (int c0 = 0; c0 < DIMC; c0 += 4) {
        const int ca = c0 + 2 * half;                 // ISA 16x4 f32 A layout: lanes>=16 hold K+2
        v2f a, b;
        float t0 = fmaf(lx, P[ca],     fmaf(ly, Q[ca],     R[ca]));
        float t1 = fmaf(lx, P[ca + 1], fmaf(ly, Q[ca + 1], R[ca + 1]));
        a.x = t0 > 0.0f ? t0 : 0.0f;                  // relu(BN(loc@W1+b1)) on the fly
        a.y = t1 > 0.0f ? t1 : 0.0f;
        b.x = lds_w2[ca * 16 + m];                    // B[K][n], mirrored K striping
        b.y = lds_w2[(ca + 1) * 16 + m];
        acc = __builtin_amdgcn_wmma_f32_16x16x4_f32(
            /*neg_a=*/false, a, /*neg_b=*/false, b,
            /*c_mod=*/(short)0, acc, /*reuse_a=*/false, /*reuse_b=*/false);
    }

    // D layout: element v -> row rowBase + v + 8*half, column = col
    const long outRowBase = posBase + ((long)k * N + rowBase + 8 * half) * DIMC + col;
#pragma unroll
    for (int v = 0; v < 8; ++v)
        out[outRowBase + (long)v * DIMC] = acc[v];
}

// ---------------------------------------------------------------------------
extern "C" void kernel_launch(void* const* d_in, const int* in_sizes, int n_in,
                              void* d_out, int out_size, void* d_ws, size_t ws_size,
                              hipStream_t stream)
{
    const float* feats  = (const float*)d_in[0];
    const int*   coords = (const int*)d_in[1];
    const float* w1     = (const float*)d_in[2];
    const float* b1     = (const float*)d_in[3];
    const float* gamma  = (const float*)d_in[4];
    const float* beta   = (const float*)d_in[5];
    const float* w2     = (const float*)d_in[6];
    const float* b2     = (const float*)d_in[7];
    float* out = (float*)d_out;

    const int N    = in_sizes[0] / DIMC;     // 60000
    const int nblk = (N + 255) / 256;

    // workspace layout (4-byte units), ~17 MB total
    float* ws = (float*)d_ws;
    size_t o = 0;
    float* locs  = ws + o; o += (size_t)4 * N;          // [2][N][2]
    float* part  = ws + o; o += (size_t)10 * nblk;      // moment partials
    float* statf = ws + o; o += 10;                     // final moments
    float* pqr   = ws + o; o += (size_t)8 * 3 * DIMC;   // folded affine params
    int* cnt0 = (int*)(ws + o); o += NW0;
    int* off0 = (int*)(ws + o); o += NW0;
    int* cnt1 = (int*)(ws + o); o += NW1;
    int* off1 = (int*)(ws + o); o += NW1;
    int* S    = (int*)(ws + o); o += 2;
    int* c0y  = (int*)(ws + o); o += (size_t)NW0 * MV0;
    int* c0x  = (int*)(ws + o); o += (size_t)NW0 * MV0;
    int* c1y  = (int*)(ws + o); o += (size_t)NW1 * MV1;
    int* c1x  = (int*)(ws + o); o += (size_t)NW1 * MV1;

    // pass-through outputs: voxel_features, voxel_coords (bit-copy)
    (void)hipMemcpyAsync(out, feats, (size_t)N * DIMC * sizeof(float),
                         hipMemcpyDeviceToDevice, stream);
    (void)hipMemcpyAsync(out + (size_t)N * DIMC, coords, (size_t)N * 4 * sizeof(int),
                         hipMemcpyDeviceToDevice, stream);

    // per-call init: counters -> 0, window cells -> -1
    (void)hipMemsetAsync(cnt0, 0, (size_t)(2 * NW0 + 2 * NW1 + 2) * sizeof(int), stream);
    (void)hipMemsetAsync(c0y, 0xFF,
                         ((size_t)2 * NW0 * MV0 + (size_t)2 * NW1 * MV1) * sizeof(int), stream);

    k_win<<<(N + 255) / 256, 256, 0, stream>>>(coords, locs, cnt0, cnt1, c0y, c0x, c1y, c1x, N);
    k_stats_part<<<nblk, 256, 0, stream>>>(locs, part, N, nblk);
    k_stats_final<<<1, 32, 0, stream>>>(part, statf, nblk);
    k_params<<<(8 * DIMC + 255) / 256, 256, 0, stream>>>(statf, w1, b1, gamma, beta, pqr, N);
    k_scan<<<1, 32, 0, stream>>>(cnt0, off0, cnt1, off1, S);
    k_compact<<<(2 * NW0 + 2 * NW1 + 255) / 256, 256, 0, stream>>>(c0y, c0x, c1y, c1x);

    const long base0 = (long)N * DIMC + (long)N * 4;
    k_emit<<<NW0, 4 * SETSZ, 0, stream>>>(cnt0, off0, c0y, c0x, S, out, base0, MV0, 0);
    k_emit<<<NW1, 16 * SETSZ, 0, stream>>>(cnt1, off1, c1y, c1x, S, out, base0, MV1, 1);

    // pos_embeds are last in the flat output: base derivable from out_size
    const long posBase = (long)out_size - (long)8 * N * DIMC;
    const int rowChunks = (N / 16 + 7) / 8;
    dim3 g(rowChunks, DIMC / 16, 8);
    k_posembed<<<g, 256, 0, stream>>>(locs, pqr, w2, b2, out, posBase, N);
}